// NeuromorphicSpikingMoE_5291399709238
// MI455X (gfx1250) — compile-verified
//
#include <hip/hip_runtime.h>

// ---------------------------------------------------------------------------
// NeuromorphicSpikingMoE on gfx1250 (MI455X)
//   Phase 0: transpose W[2048,64] -> BT[64,2048]           (parallel, trivial)
//   Phase 1: currents = hidden @ W via V_WMMA_F32_16X16X4  (parallel, WMMA)
//   Phase 2: serial LIF recursion, stores pre-reset mp     (1 wave, in-place)
//   Phase 3: softmax routing from stored mp                (parallel)
// ---------------------------------------------------------------------------

typedef float v2f __attribute__((ext_vector_type(2)));
typedef float v8f __attribute__((ext_vector_type(8)));

#define HIDDEN 2048
#define EXPERTS 64
#define TSTEPS 8192   // 4 * 2048 rows, scanned in b-major order

// ---------------------------------------------------------------------------
// Phase 0: BT[n*2048 + k] = W[k*64 + n]
// ---------------------------------------------------------------------------
__global__ __launch_bounds__(256) void transpose_w_kernel(const float* __restrict__ W,
                                                          float* __restrict__ BT) {
    int idx = blockIdx.x * 256 + threadIdx.x;      // 2048*64 = 131072 elements
    int k = idx >> 6;
    int n = idx & 63;
    BT[(size_t)n * HIDDEN + k] = W[(size_t)k * EXPERTS + n];
}

// ---------------------------------------------------------------------------
// Phase 1: C[8192,64] = A[8192,2048] * W[2048,64] using f32 WMMA 16x16x4.
// One wave per 16-row M-tile; each wave accumulates all four 16-col N-tiles.
// A frag (16x4):  lanes 0-15 -> M=lane,   K=k0+{0,1}; lanes 16-31 -> K=k0+2+{0,1}
// B frag (4x16):  lanes 0-15 -> N=lane,   K=k0+{0,1}; lanes 16-31 -> K=k0+2+{0,1}
//                 (contiguous in BT since BT is [N][K])
// C frag (16x16): lanes 0-15 -> N=lane, M=vgpr r; lanes 16-31 -> N=lane-16, M=r+8
// ---------------------------------------------------------------------------
__global__ __launch_bounds__(256) void gemm_wmma_kernel(const float* __restrict__ A,
                                                        const float* __restrict__ BT,
                                                        float* __restrict__ C) {
    const int wave  = threadIdx.x >> 5;
    const int lane  = threadIdx.x & 31;
    const int mtile = blockIdx.x * 8 + wave;       // 512 M-tiles total
    const int m0    = mtile * 16;
    const int half  = lane >> 4;                   // which K pair
    const int sub   = lane & 15;                   // M row (A) / N col (B)

    const float* arow = A  + (size_t)(m0 + sub) * HIDDEN + half * 2;
    const float* b0p  = BT + (size_t)(0 * 16 + sub) * HIDDEN + half * 2;
    const float* b1p  = BT + (size_t)(1 * 16 + sub) * HIDDEN + half * 2;
    const float* b2p  = BT + (size_t)(2 * 16 + sub) * HIDDEN + half * 2;
    const float* b3p  = BT + (size_t)(3 * 16 + sub) * HIDDEN + half * 2;

    v8f acc0 = {}, acc1 = {}, acc2 = {}, acc3 = {};

    for (int k0 = 0; k0 < HIDDEN; k0 += 4) {
        // locality=3 -> WGP-scope prefetch (pulls into all cache levels; the
        // SYS-scope variant from locality 0 only warms GL2, useless for a
        // stream consumed by the same wave a few hundred cycles later).
        __builtin_prefetch(arow + k0 + 512, 0, 3);
        v2f a  = *(const v2f*)(arow + k0);
        v2f b0 = *(const v2f*)(b0p + k0);
        v2f b1 = *(const v2f*)(b1p + k0);
        v2f b2 = *(const v2f*)(b2p + k0);
        v2f b3 = *(const v2f*)(b3p + k0);
        acc0 = __builtin_amdgcn_wmma_f32_16x16x4_f32(false, a, false, b0, (short)0, acc0, false, false);
        acc1 = __builtin_amdgcn_wmma_f32_16x16x4_f32(false, a, false, b1, (short)0, acc1, false, false);
        acc2 = __builtin_amdgcn_wmma_f32_16x16x4_f32(false, a, false, b2, (short)0, acc2, false, false);
        acc3 = __builtin_amdgcn_wmma_f32_16x16x4_f32(false, a, false, b3, (short)0, acc3, false, false);
    }

    // Store C tile: row = m0 + half*8 + r, col = ntile*16 + sub
#pragma unroll
    for (int r = 0; r < 8; ++r) {
        size_t row = (size_t)(m0 + half * 8 + r) * EXPERTS;
        C[row +  0 + sub] = acc0[r];
        C[row + 16 + sub] = acc1[r];
        C[row + 32 + sub] = acc2[r];
        C[row + 48 + sub] = acc3[r];
    }
}

// ---------------------------------------------------------------------------
// Phase 2: serial LIF recursion, single wave, experts {2*lane, 2*lane+1}.
// Reads currents[t], writes pre-reset membrane potential back in place.
// Double-buffered 8-step chunks to hide load latency behind the recursion.
// ---------------------------------------------------------------------------
#define CH 8

__device__ __forceinline__ void lif_step(float2 cur, float2& mp, float2& rf,
                                         float2* __restrict__ out) {
    // mp = mp*leak + cur*active*dt  (active checked on incoming refractory)
    mp.x *= 0.9f;
    mp.y *= 0.9f;
    float a0 = (rf.x <= 0.0f) ? 0.1f : 0.0f;
    float a1 = (rf.y <= 0.0f) ? 0.1f : 0.0f;
    mp.x += cur.x * a0;
    mp.y += cur.y * a1;
    rf.x = fmaxf(rf.x - 0.1f, 0.0f);
    rf.y = fmaxf(rf.y - 0.1f, 0.0f);
    *out = mp;                                  // pre-reset mp (spike recomputable)
    bool s0 = mp.x > 1.0f;
    bool s1 = mp.y > 1.0f;
    mp.x = s0 ? 0.0f : mp.x;
    mp.y = s1 ? 0.0f : mp.y;
    rf.x = s0 ? 1.0f : rf.x;
    rf.y = s1 ? 1.0f : rf.y;
}

__global__ __launch_bounds__(32) void lif_scan_kernel(float2* __restrict__ buf) {
    const int lane = threadIdx.x;
    float2 mp = {0.0f, 0.0f};
    float2 rf = {0.0f, 0.0f};

    float2 bufA[CH], bufB[CH];
#pragma unroll
    for (int i = 0; i < CH; ++i) bufA[i] = buf[(size_t)i * 32 + lane];

    for (int t0 = 0; t0 < TSTEPS; t0 += 2 * CH) {
        // prefetch chunk B while computing chunk A
#pragma unroll
        for (int i = 0; i < CH; ++i) bufB[i] = buf[(size_t)(t0 + CH + i) * 32 + lane];
#pragma unroll
        for (int i = 0; i < CH; ++i)
            lif_step(bufA[i], mp, rf, &buf[(size_t)(t0 + i) * 32 + lane]);

        // prefetch next chunk A while computing chunk B
        if (t0 + 2 * CH < TSTEPS) {
#pragma unroll
            for (int i = 0; i < CH; ++i) bufA[i] = buf[(size_t)(t0 + 2 * CH + i) * 32 + lane];
        }
#pragma unroll
        for (int i = 0; i < CH; ++i)
            lif_step(bufB[i], mp, rf, &buf[(size_t)(t0 + CH + i) * 32 + lane]);
    }
}

// ---------------------------------------------------------------------------
// Phase 3: routing = softmax(any_spike ? one_hot(spike) : mp), per time step.
// One wave per row; wave32 shfl_xor reductions; jax-style max-subtracted softmax.
// ---------------------------------------------------------------------------
__global__ __launch_bounds__(256) void routing_kernel(const float2* __restrict__ mpbuf,
                                                      float2* __restrict__ out) {
    const int row  = blockIdx.x * 8 + (threadIdx.x >> 5);
    const int lane = threadIdx.x & 31;

    float2 mp = mpbuf[(size_t)row * 32 + lane];
    bool s0 = mp.x > 1.0f;
    bool s1 = mp.y > 1.0f;

    unsigned long long bal = __ballot(s0 || s1);
    bool any = (bal != 0ull);

    float x0 = any ? (s0 ? 1.0f : 0.0f) : mp.x;   // threshold == 1.0 so mp/thr == mp
    float x1 = any ? (s1 ? 1.0f : 0.0f) : mp.y;

    float m = fmaxf(x0, x1);
#pragma unroll
    for (int off = 16; off >= 1; off >>= 1)
        m = fmaxf(m, __shfl_xor(m, off, 32));

    float e0 = __expf(x0 - m);
    float e1 = __expf(x1 - m);
    float s  = e0 + e1;
#pragma unroll
    for (int off = 16; off >= 1; off >>= 1)
        s += __shfl_xor(s, off, 32);

    float inv = 1.0f / s;
    float2 r;
    r.x = e0 * inv;
    r.y = e1 * inv;
    out[(size_t)row * 32 + lane] = r;
}

// ---------------------------------------------------------------------------
extern "C" void kernel_launch(void* const* d_in, const int* in_sizes, int n_in,
                              void* d_out, int out_size, void* d_ws, size_t ws_size,
                              hipStream_t stream) {
    const float* hidden = (const float*)d_in[0];   // [4,2048,2048] f32
    const float* W      = (const float*)d_in[1];   // [2048,64] f32
    float* out          = (float*)d_out;           // [4,2048,64] f32

    // workspace: [0, 2MB) currents -> pre-reset mp (in place); [2MB, 2.5MB) BT
    float* curr = (float*)d_ws;                                   // 8192*64 f32
    float* BT   = (float*)((char*)d_ws + (size_t)TSTEPS * EXPERTS * sizeof(float));

    // Phase 0: transpose weights
    transpose_w_kernel<<<(HIDDEN * EXPERTS) / 256, 256, 0, stream>>>(W, BT);

    // Phase 1: currents = hidden @ W   (512 M-tiles, 8 waves/block)
    gemm_wmma_kernel<<<64, 256, 0, stream>>>(hidden, BT, curr);

    // Phase 2: serial recursion (single wave), in-place mp
    lif_scan_kernel<<<1, 32, 0, stream>>>((float2*)curr);

    // Phase 3: parallel softmax routing
    routing_kernel<<<TSTEPS / 8, 256, 0, stream>>>((const float2*)curr, (float2*)out);
}